// DeformAttn_88811333747443
// MI455X (gfx1250) — compile-verified
//
#include <hip/hip_runtime.h>
#include <hip/hip_bf16.h>
#include <math.h>

// ---------------------------------------------------------------------------
// Deformable attention block for MI455X (gfx1250, wave32).
//  - Channel GEMMs (q/k/v projections + FFN) via V_WMMA_F32_16X16X4_F32.
//  - W^T tile staged in LDS: cooperative b128->ds_store (authoritative data)
//    plus a Tensor Data Mover load of the same tile (TENSORcnt-tracked DMA);
//    A-operand then comes from ds_load with 32-bit addressing, shared by all
//    4 waves of the workgroup.
//  - X stream prefetched ahead with global_prefetch.
//  - All intermediates staged in d_ws (~85 MB) => resident in the 192MB L2.
//  - Deformable bilinear gather + softmax: one lane per (b, pixel, head).
// ---------------------------------------------------------------------------

typedef float v2f __attribute__((ext_vector_type(2)));
typedef float v8f __attribute__((ext_vector_type(8)));
typedef unsigned int u32x4 __attribute__((ext_vector_type(4)));
typedef int i32x4 __attribute__((ext_vector_type(4)));
typedef int i32x8 __attribute__((ext_vector_type(8)));

#define DA_H   96
#define DA_W   96
#define DA_HW  (DA_H * DA_W)       // 9216
#define DA_C   144
#define DA_C2  288
#define DA_HEADS 12
#define DA_DH  12
#define DA_T   2
#define DA_B   2
#define DA_K   9

// ---------------------------------------------------------------------------
// GEMM over channel dim:  Y[n][dout][p] = act( sum_c W[c][dout]*X[n][c][p] + b[dout] ) (+ res)
// One wave computes one 16x16 (dout x pixel) tile via wmma_f32_16x16x4_f32.
// blockDim.x = 128 (4 waves); grid = (P/64, Cout/16, nbatch).
// LDS holds the W^T tile lw[c][j] = W[c*Cout + db + j], c in [0,Cin), j in [0,16).
// ---------------------------------------------------------------------------
__global__ __launch_bounds__(128)
void gemm_ch_kernel(const float* __restrict__ X, const float* __restrict__ Wm,
                    const float* __restrict__ bias, const float* __restrict__ res,
                    float* __restrict__ Y, int Cin, int Cout, int P, int act)
{
    __shared__ float lw[DA_C2 * 16];           // up to 288 x 16 fp32 = 18 KB

    const int wave = threadIdx.x >> 5;
    const int lane = threadIdx.x & 31;
    const int nt   = blockIdx.x * 4 + wave;   // pixel tile
    const int mt   = blockIdx.y;              // out-channel tile
    const int bb   = blockIdx.z;              // batch (b*t)

    const float* Xb = X + (size_t)bb * Cin * P;
    float*       Yb = Y + (size_t)bb * Cout * P;
    const float* Rb = res ? (res + (size_t)bb * Cout * P) : nullptr;

    const int pb = nt * 16;
    const int db = mt * 16;
    const int m  = lane & 15;     // row-in-tile for A, col-in-tile for B
    const int hh = lane >> 4;     // lane half selects K pair

    // ---- stage W^T tile (Cin rows x 16 floats, row stride Cout) into LDS ----
    // (a) Tensor Data Mover DMA of the tile (async, TENSORcnt-tracked).
#if defined(__gfx1250__) && __has_builtin(__builtin_amdgcn_tensor_load_to_lds)
    if (threadIdx.x < 32) {       // one TDM issue per workgroup (wave 0)
        const unsigned long long ga =
            (unsigned long long)(uintptr_t)(Wm + (size_t)db);
        u32x4 g0 = {};
        i32x8 g1 = {};
        i32x4 g2 = {};
        i32x4 g3 = {};
        // D# group0: count=1 | lds_addr (lw is the sole LDS alloc -> offset 0)
        //            | 57-bit global byte address | type=2 ("image")
        g0[0] = 1u;
        g0[1] = 0u;
        g0[2] = (unsigned)(ga & 0xFFFFFFFFu);
        g0[3] = ((unsigned)(ga >> 32) & 0x01FFFFFFu) | (2u << 30);
        // D# group1: data_size=2 (4B); tensor_dim0=Cout; tensor_dim1=Cin;
        //            tile_dim0=16; tile_dim1=Cin; tensor_dim0_stride=Cout
        g1[0] = 2 << 16;                      // data_size = 4 bytes
        g1[1] = (Cout & 0xFFFF) << 16;        // tensor_dim0[15:0] @ bits 63:48
        g1[2] = (Cin  & 0xFFFF) << 16;        // tensor_dim1[15:0] @ bits 95:80
        g1[3] = 16 << 16;                     // tile_dim0 @ bits 127:112
        g1[4] = Cin & 0xFFFF;                 // tile_dim1 @ bits 143:128
        g1[5] = Cout;                         // tensor_dim0_stride[31:0]
        g1[6] = 0;
        g1[7] = 0;
#if __clang_major__ >= 23
        i32x8 g4 = {};
        __builtin_amdgcn_tensor_load_to_lds(g0, g1, g2, g3, g4, 0);
#else
        __builtin_amdgcn_tensor_load_to_lds(g0, g1, g2, g3, 0);
#endif
        __builtin_amdgcn_s_wait_tensorcnt(0);
    }
#endif
    // (b) Cooperative load of the same bytes (authoritative; also keeps the
    //     LDS reads below alive for alias analysis — the TDM's LDS target is
    //     opaque to the compiler).
    for (int c = (int)(threadIdx.x >> 2); c < Cin; c += 32) {
        const int j4 = ((int)threadIdx.x & 3) * 4;
        const float4 wv = *(const float4*)&Wm[(size_t)c * Cout + db + j4];
        *(float4*)&lw[c * 16 + j4] = wv;
    }
    __syncthreads();

    // ---- K loop: pointer-increment addressing, 4 channels per WMMA step ----
    const float* xp0 = Xb + (size_t)(2 * hh) * P + pb + m;   // row c0
    const float* xp1 = xp0 + P;                              // row c0+1
    const float* lp  = lw + (2 * hh) * 16 + m;               // LDS A-frag ptr
    v8f acc = {};
    for (int cb = 0; cb < Cin; cb += 4) {
        v2f a, bq;
        // A = W^T tile from LDS (two ds_loads 64B apart -> ds_load_2addr)
        a.x  = lp[0];
        a.y  = lp[16];
        // B = X tile from global (coalesced b32 across lanes 0..15 / 16..31)
        bq.x = xp0[0];
        bq.y = xp1[0];
        // Prefetch the X stream 8 channels ahead (speculative; OOB dropped).
        __builtin_prefetch(xp0 + (size_t)8 * P, 0, 3);
        acc = __builtin_amdgcn_wmma_f32_16x16x4_f32(
                  /*neg_a=*/false, a, /*neg_b=*/false, bq,
                  /*c_mod=*/(short)0, acc, /*reuse_a=*/false, /*reuse_b=*/false);
        lp  += 4 * 16;
        xp0 += (size_t)4 * P;
        xp1 += (size_t)4 * P;
    }

#pragma unroll
    for (int r = 0; r < 8; ++r) {
        const int row = db + r + 8 * hh;              // C/D layout: M = r + 8*(lane>=16)
        float v = acc[r] + bias[row];
        if (act) {                                    // exact GELU
            v = 0.5f * v * (1.0f + erff(v * 0.70710678118654752f));
        }
        if (Rb) v += Rb[(size_t)row * P + pb + m];
        Yb[(size_t)row * P + pb + m] = v;
    }
}

// ---------------------------------------------------------------------------
// Deformable attention: one lane per (b, head, pixel).
// ---------------------------------------------------------------------------
__device__ __forceinline__ float da_sample(const float* __restrict__ ch, int y, int x)
{
    const bool valid = (y >= 0) & (y < DA_H) & (x >= 0) & (x < DA_W);
    int yc = y < 0 ? 0 : (y > DA_H - 1 ? DA_H - 1 : y);
    int xc = x < 0 ? 0 : (x > DA_W - 1 ? DA_W - 1 : x);
    return valid ? ch[yc * DA_W + xc] : 0.0f;
}

__global__ __launch_bounds__(256)
void deform_attn_kernel(const float* __restrict__ qp, const float* __restrict__ kp,
                        const float* __restrict__ vp, const float* __restrict__ off,
                        float* __restrict__ xout)
{
    const int tid = blockIdx.x * blockDim.x + threadIdx.x;
    if (tid >= DA_B * DA_HEADS * DA_HW) return;

    const int p = tid % DA_HW;
    const int h = (tid / DA_HW) % DA_HEADS;
    const int b = tid / (DA_HEADS * DA_HW);
    const int py = p / DA_W;
    const int px = p % DA_W;

    // q, scaled by dh^-0.5
    float qv[DA_DH];
#pragma unroll
    for (int d = 0; d < DA_DH; ++d)
        qv[d] = qp[((size_t)b * DA_C + h * DA_DH + d) * DA_HW + p] * 0.2886751345948129f;

    const int gk = h >> 1;          // sampling group for K channels (0..5)
    const int gv = 6 + (h >> 1);    // sampling group for V channels (6..11)

    // ---- pass 1: scores over T*K = 18 keys ----
    float sc[DA_T * DA_K];
    for (int t = 0; t < DA_T; ++t) {
        const float* kpb  = kp  + (size_t)(b * DA_T + t) * DA_C * DA_HW;
        const float* offb = off + (size_t)(b * DA_T + t) * (12 * DA_K * 2) * DA_HW;
        for (int k = 0; k < DA_K; ++k) {
            const float dy = offb[(size_t)((gk * DA_K + k) * 2 + 0) * DA_HW + p];
            const float dx = offb[(size_t)((gk * DA_K + k) * 2 + 1) * DA_HW + p];
            const float ys = dy + (float)(k / 3 + py - 1);
            const float xs = dx + (float)(k % 3 + px - 1);
            const float y0 = floorf(ys), x0 = floorf(xs);
            const float ly = ys - y0,   lx = xs - x0;
            const int y0i = (int)y0,    x0i = (int)x0;
            const float w00 = (1.f - ly) * (1.f - lx), w01 = (1.f - ly) * lx;
            const float w10 = ly * (1.f - lx),         w11 = ly * lx;
            float s = 0.f;
#pragma unroll
            for (int d = 0; d < DA_DH; ++d) {
                const float* ch = kpb + (size_t)(h * DA_DH + d) * DA_HW;
                const float val = w00 * da_sample(ch, y0i,     x0i)
                                + w01 * da_sample(ch, y0i,     x0i + 1)
                                + w10 * da_sample(ch, y0i + 1, x0i)
                                + w11 * da_sample(ch, y0i + 1, x0i + 1);
                s += qv[d] * val;
            }
            sc[t * DA_K + k] = s;
        }
    }

    // ---- softmax over 18 ----
    float mx = sc[0];
#pragma unroll
    for (int i = 1; i < DA_T * DA_K; ++i) mx = fmaxf(mx, sc[i]);
    float sum = 0.f;
#pragma unroll
    for (int i = 0; i < DA_T * DA_K; ++i) { sc[i] = expf(sc[i] - mx); sum += sc[i]; }
    const float inv = 1.f / sum;

    // ---- pass 2: weighted V gather (V uses its own offset group) ----
    float outv[DA_DH];
#pragma unroll
    for (int d = 0; d < DA_DH; ++d) outv[d] = 0.f;

    for (int t = 0; t < DA_T; ++t) {
        const float* vpb  = vp  + (size_t)(b * DA_T + t) * DA_C * DA_HW;
        const float* offb = off + (size_t)(b * DA_T + t) * (12 * DA_K * 2) * DA_HW;
        for (int k = 0; k < DA_K; ++k) {
            const float dy = offb[(size_t)((gv * DA_K + k) * 2 + 0) * DA_HW + p];
            const float dx = offb[(size_t)((gv * DA_K + k) * 2 + 1) * DA_HW + p];
            const float ys = dy + (float)(k / 3 + py - 1);
            const float xs = dx + (float)(k % 3 + px - 1);
            const float y0 = floorf(ys), x0 = floorf(xs);
            const float ly = ys - y0,   lx = xs - x0;
            const int y0i = (int)y0,    x0i = (int)x0;
            const float w00 = (1.f - ly) * (1.f - lx), w01 = (1.f - ly) * lx;
            const float w10 = ly * (1.f - lx),         w11 = ly * lx;
            const float aw = sc[t * DA_K + k] * inv;
#pragma unroll
            for (int d = 0; d < DA_DH; ++d) {
                const float* ch = vpb + (size_t)(h * DA_DH + d) * DA_HW;
                const float val = w00 * da_sample(ch, y0i,     x0i)
                                + w01 * da_sample(ch, y0i,     x0i + 1)
                                + w10 * da_sample(ch, y0i + 1, x0i)
                                + w11 * da_sample(ch, y0i + 1, x0i + 1);
                outv[d] += aw * val;
            }
        }
    }

#pragma unroll
    for (int d = 0; d < DA_DH; ++d)
        xout[((size_t)b * DA_C + h * DA_DH + d) * DA_HW + p] = outv[d];
}

// ---------------------------------------------------------------------------
// Launch: 3 WMMA projections -> deformable attention -> WMMA FFN (GELU) ->
// WMMA FFN2 + residual into d_out. Intermediates live in d_ws (L2-resident).
// ---------------------------------------------------------------------------
extern "C" void kernel_launch(void* const* d_in, const int* in_sizes, int n_in,
                              void* d_out, int out_size, void* d_ws, size_t ws_size,
                              hipStream_t stream)
{
    const float* q   = (const float*)d_in[0];
    const float* k   = (const float*)d_in[1];
    const float* v   = (const float*)d_in[2];
    const float* off = (const float*)d_in[3];
    const float* Wq  = (const float*)d_in[4];
    const float* bq  = (const float*)d_in[5];
    const float* Wk  = (const float*)d_in[6];
    const float* bk  = (const float*)d_in[7];
    const float* Wv  = (const float*)d_in[8];
    const float* bv  = (const float*)d_in[9];
    const float* W1  = (const float*)d_in[10];
    const float* b1  = (const float*)d_in[11];
    const float* W2  = (const float*)d_in[12];
    const float* b2  = (const float*)d_in[13];

    float* ws   = (float*)d_ws;
    const size_t HW = (size_t)DA_HW;
    float* qp   = ws;                              // B   * 144 * HW
    float* kp   = qp  + (size_t)DA_B * DA_C * HW;  // B*T * 144 * HW
    float* vp   = kp  + (size_t)DA_B * DA_T * DA_C * HW;
    float* xat  = vp  + (size_t)DA_B * DA_T * DA_C * HW;   // B * 144 * HW
    float* hmid = xat + (size_t)DA_B * DA_C * HW;          // B * 288 * HW

    const dim3 blk(128);
    const int ntx = DA_HW / 16 / 4;   // 144 blocks of 4 waves -> 576 pixel tiles

    // Projections (fp32 WMMA GEMMs)
    gemm_ch_kernel<<<dim3(ntx, DA_C / 16, DA_B),          blk, 0, stream>>>(
        q, Wq, bq, nullptr, qp, DA_C, DA_C, DA_HW, 0);
    gemm_ch_kernel<<<dim3(ntx, DA_C / 16, DA_B * DA_T),   blk, 0, stream>>>(
        k, Wk, bk, nullptr, kp, DA_C, DA_C, DA_HW, 0);
    gemm_ch_kernel<<<dim3(ntx, DA_C / 16, DA_B * DA_T),   blk, 0, stream>>>(
        v, Wv, bv, nullptr, vp, DA_C, DA_C, DA_HW, 0);

    // Deformable attention (gather + softmax), lane per (b, head, pixel)
    const int nthr = DA_B * DA_HEADS * DA_HW;
    deform_attn_kernel<<<(nthr + 255) / 256, 256, 0, stream>>>(qp, kp, vp, off, xat);

    // FFN: 144 -> 288 with exact GELU, then 288 -> 144 + residual(xat) into d_out
    gemm_ch_kernel<<<dim3(ntx, DA_C2 / 16, DA_B), blk, 0, stream>>>(
        xat, W1, b1, nullptr, hmid, DA_C, DA_C2, DA_HW, 1);
    gemm_ch_kernel<<<dim3(ntx, DA_C / 16, DA_B),  blk, 0, stream>>>(
        hmid, W2, b2, xat, (float*)d_out, DA_C2, DA_C, DA_HW, 0);
}